// MessageGraphConvolution_30494267802264
// MI455X (gfx1250) — compile-verified
//
#include <hip/hip_runtime.h>

typedef __attribute__((ext_vector_type(2))) float v2f;
typedef __attribute__((ext_vector_type(8))) float v8f;

#define IN_F 256
#define OUT_F 256

// ---------------------------------------------------------------------------
// Zero the workspace (agg + counts). Harness poisons ws with 0xAA, so we must
// clear it ourselves on every launch.
// ---------------------------------------------------------------------------
__global__ void zero_ws_kernel(float* __restrict__ ws, int n) {
    int i = blockIdx.x * blockDim.x + threadIdx.x;
    if (i < n) ws[i] = 0.0f;
}

// ---------------------------------------------------------------------------
// Scatter-mean phase: one 256-thread block per edge.
// src/dst loads are block-uniform (scalarized), x[src] gather is coalesced.
// agg[dst, f] += x[src, f]  (f32 atomics, land in L2)
// cnt[dst]    += 1          (lane 0 only)
// ---------------------------------------------------------------------------
__global__ __launch_bounds__(IN_F) void scatter_mean_kernel(
        const float* __restrict__ x,
        const int*   __restrict__ ei,   // edge_index flat: [0][e]=src, [1][e]=dst
        float* __restrict__ agg,
        float* __restrict__ cnt,
        int num_edges) {
    const int e = blockIdx.x;
    const int f = threadIdx.x;
    const int src = ei[e];
    const int dst = ei[num_edges + e];
    atomicAdd(&agg[(long)dst * IN_F + f], x[(long)src * IN_F + f]);
    if (f == 0) atomicAdd(&cnt[dst], 1.0f);
}

// ---------------------------------------------------------------------------
// agg[i] /= max(cnt[row], 1)
// ---------------------------------------------------------------------------
__global__ void normalize_kernel(float* __restrict__ agg,
                                 const float* __restrict__ cnt,
                                 int total) {
    int i = blockIdx.x * blockDim.x + threadIdx.x;
    if (i < total) {
        float c = cnt[i / IN_F];
        c = fmaxf(c, 1.0f);
        agg[i] = agg[i] / c;
    }
}

// ---------------------------------------------------------------------------
// Fused GEMM: out = agg @ W^T + x @ B^T  using V_WMMA_F32_16X16X4_F32.
// Grid: one block per 16-row M-tile; 4 waves/block, each wave owns a
// 16(M) x 64(N) strip (4 accumulator tiles, A-fragment reused 4x).
//
// A fragment (16x4 f32): lane l: row = m0 + (l&15); VGPR0/1 hold K = kb,kb+1
//   where kb = k + 2*(l>>4)  -> contiguous float2 from the A row.
// B fragment (4x16 f32): Bmat[kk][nn] = W[n0+nn][k+kk]; lane l: col = n0+(l&15),
//   VGPR0/1 hold K = kb,kb+1 -> contiguous float2 from W row `col`.
// C/D (16x16 f32): VGPR v -> row m0 + v + 8*(l>>4), col = n0 + (l&15).
// ---------------------------------------------------------------------------
__global__ __launch_bounds__(128) void gemm_wmma_kernel(
        const float* __restrict__ agg,
        const float* __restrict__ x,
        const float* __restrict__ W,
        const float* __restrict__ B,
        float* __restrict__ out,
        int num_nodes) {
    const int m0   = blockIdx.x * 16;
    const int wv   = threadIdx.x >> 5;   // wave in block: 0..3
    const int lane = threadIdx.x & 31;
    const int hf   = lane >> 4;          // half-wave select
    const int l15  = lane & 15;
    const int n0   = wv * 64;

    int arow = m0 + l15;
    if (arow >= num_nodes) arow = num_nodes - 1;   // clamp (stores are guarded)
    const float* __restrict__ a1p = agg + (long)arow * IN_F;
    const float* __restrict__ a2p = x   + (long)arow * IN_F;

    v8f acc[4] = {};

    for (int k = 0; k < IN_F; k += 4) {
        const int kb = k + hf * 2;
        const v2f a1 = *(const v2f*)(a1p + kb);
        const v2f a2 = *(const v2f*)(a2p + kb);
#pragma unroll
        for (int j = 0; j < 4; ++j) {
            const int col = n0 + j * 16 + l15;
            const v2f bw = *(const v2f*)(W + (long)col * IN_F + kb);
            const v2f bb = *(const v2f*)(B + (long)col * IN_F + kb);
            // acc = agg_frag * W_frag + acc
            acc[j] = __builtin_amdgcn_wmma_f32_16x16x4_f32(
                false, a1, false, bw, (short)0, acc[j], false, false);
            // acc = x_frag * B_frag + acc   (accumulator chaining, no D->A/B hazard)
            acc[j] = __builtin_amdgcn_wmma_f32_16x16x4_f32(
                false, a2, false, bb, (short)0, acc[j], false, false);
        }
    }

#pragma unroll
    for (int j = 0; j < 4; ++j) {
        const int col = n0 + j * 16 + l15;
#pragma unroll
        for (int v = 0; v < 8; ++v) {
            const int r = m0 + v + hf * 8;
            if (r < num_nodes) out[(long)r * OUT_F + col] = acc[j][v];
        }
    }
}

// ---------------------------------------------------------------------------
// Host launcher
// ---------------------------------------------------------------------------
extern "C" void kernel_launch(void* const* d_in, const int* in_sizes, int n_in,
                              void* d_out, int out_size, void* d_ws, size_t ws_size,
                              hipStream_t stream) {
    const float* x  = (const float*)d_in[0];
    const int*   ei = (const int*)d_in[1];
    const float* W  = (const float*)d_in[2];
    const float* B  = (const float*)d_in[3];
    float* out = (float*)d_out;

    const int num_nodes = in_sizes[0] / IN_F;
    const int num_edges = in_sizes[1] / 2;

    float* agg = (float*)d_ws;                       // [num_nodes * IN_F]
    float* cnt = agg + (long)num_nodes * IN_F;       // [num_nodes]

    // 1) clear workspace
    const int zn = num_nodes * IN_F + num_nodes;
    zero_ws_kernel<<<(zn + 255) / 256, 256, 0, stream>>>((float*)d_ws, zn);

    // 2) scatter-add messages + counts
    scatter_mean_kernel<<<num_edges, IN_F, 0, stream>>>(x, ei, agg, cnt, num_edges);

    // 3) mean normalization
    const int nn = num_nodes * IN_F;
    normalize_kernel<<<(nn + 255) / 256, 256, 0, stream>>>(agg, cnt, nn);

    // 4) fused dual GEMM on the f32 WMMA pipe
    const int mtiles = (num_nodes + 15) / 16;
    gemm_wmma_kernel<<<mtiles, 128, 0, stream>>>(agg, x, W, B, out, num_nodes);
}